// VariationalFFNDynamicStable_12996571038354
// MI455X (gfx1250) — compile-verified
//
#include <hip/hip_runtime.h>

#define Bb 2
#define Nn 256
#define Kk 16
#define BN (Bb * Nn)
#define ALPHA 0.01f
#define LAM 1.0f
#define KAPPA 1.0f
#define EPSF 1e-6f
#define GCLIP 1000.0f

typedef __attribute__((ext_vector_type(2))) float v2f;
typedef __attribute__((ext_vector_type(8))) float v8f;

// ---------------------------------------------------------------------------
// Kernel 1: E_n = expm(sum_a phi[n,a] * G_a), one block (256 thr) per node.
// Scaling-and-squaring: X/16, Taylor order 8 (Horner), then square 4 times.
// ---------------------------------------------------------------------------
__global__ __launch_bounds__(256) void vfe_expm_kernel(
    const float* __restrict__ phi, const float* __restrict__ gen,
    float* __restrict__ E) {
  const int n = blockIdx.x;           // 0..BN-1
  const int tid = threadIdx.x;
  const int i = tid >> 4, j = tid & 15;

  __shared__ float sX[16][17];
  __shared__ float sR[16][17];

  const float p0 = phi[n * 3 + 0];
  const float p1 = phi[n * 3 + 1];
  const float p2 = phi[n * 3 + 2];
  float x = p0 * gen[0 * 256 + i * 16 + j] +
            p1 * gen[1 * 256 + i * 16 + j] +
            p2 * gen[2 * 256 + i * 16 + j];
  x *= 0.0625f;  // 2^-4 scaling
  sX[i][j] = x;
  sR[i][j] = (i == j) ? 1.0f : 0.0f;
  __syncthreads();

  // Horner: R = I + Xs*(I + Xs/2*(I + ... Xs/8))
  #pragma unroll
  for (int k = 8; k >= 1; --k) {
    float acc = 0.0f;
    #pragma unroll
    for (int m = 0; m < 16; ++m) acc += sX[i][m] * sR[m][j];
    __syncthreads();
    sR[i][j] = ((i == j) ? 1.0f : 0.0f) + acc * (1.0f / (float)k);
    __syncthreads();
  }
  // 4 squarings
  #pragma unroll
  for (int s = 0; s < 4; ++s) {
    float acc = 0.0f;
    #pragma unroll
    for (int m = 0; m < 16; ++m) acc += sR[i][m] * sR[m][j];
    __syncthreads();
    sR[i][j] = acc;
    __syncthreads();
  }
  E[n * 256 + i * 16 + j] = sR[i][j];
}

// ---------------------------------------------------------------------------
// Kernel 2: per node n:
//   r = 1/(max(sigma_q,EPS)+EPS)
//   Minv = E^T diag(r) E      (16x16)
//   y    = E^T mu             (16)
//   w    = E^T (r .* mu)      (16)
//   c    = sum mu^2 r         (scalar)
// ---------------------------------------------------------------------------
__global__ __launch_bounds__(256) void vfe_prep_kernel(
    const float* __restrict__ sigma_q, const float* __restrict__ mu_q,
    const float* __restrict__ E, float* __restrict__ Minv,
    float* __restrict__ Y, float* __restrict__ W, float* __restrict__ Cc) {
  const int n = blockIdx.x;
  const int tid = threadIdx.x;
  const int k = tid >> 4, l = tid & 15;

  __shared__ float sE[16][17];
  __shared__ float sr[16], smu[16], srmu[16];

  sE[k][l] = E[n * 256 + k * 16 + l];
  if (tid < 16) {
    float sq = fmaxf(sigma_q[n * 16 + tid], EPSF);
    float rr = 1.0f / (sq + EPSF);
    float m = mu_q[n * 16 + tid];
    sr[tid] = rr;
    smu[tid] = m;
    srmu[tid] = rr * m;
  }
  __syncthreads();

  float acc = 0.0f;
  #pragma unroll
  for (int m = 0; m < 16; ++m) acc += sE[m][k] * sr[m] * sE[m][l];
  Minv[n * 256 + k * 16 + l] = acc;

  if (tid < 16) {
    float y = 0.0f, w = 0.0f;
    #pragma unroll
    for (int m = 0; m < 16; ++m) {
      y += sE[m][tid] * smu[m];
      w += sE[m][tid] * srmu[m];
    }
    Y[n * 16 + tid] = y;
    W[n * 16 + tid] = w;
  }
  if (tid == 0) {
    float c = 0.0f;
    #pragma unroll
    for (int m = 0; m < 16; ++m) c += smu[m] * srmu[m];
    Cc[n] = c;
  }
}

// ---------------------------------------------------------------------------
// Kernel 3: pairwise core. One block per (b, i-tile of 16 columns).
// 4 waves split j = 0..255, j-loop unrolled x2 for load/WMMA pipelining.
// Per (wave, j):
//   P = Minv_j @ U_tile  via 4x V_WMMA_F32_16X16X4_F32
//   q_i = y_i . P[:,i],  t_i = y_i . w_j,  kl = 0.5(q - 2t + c_j)
//   m = P[:,i] - w_j ; acc1 += beta*m ; acc2 += beta*kl*m ; s1 += beta*kl
// Output per i: h_i = LAM*acc1 + (LAM/KAPPA)*(acc2 - s1*acc1)
// (final grad contribution is E_i @ h_i, applied in kernel 4)
// C-layout: lane L (L<16) holds rows 0..7 of column L; lane L+16 rows 8..15.
// EXEC is all-ones at every WMMA (uniform trip counts, no divergence).
// ---------------------------------------------------------------------------
__global__ __launch_bounds__(128) void vfe_pairs_kernel(
    const float* __restrict__ Y, const float* __restrict__ W,
    const float* __restrict__ Cc, const float* __restrict__ Minv,
    const float* __restrict__ beta, float* __restrict__ H) {
  const int blk = blockIdx.x;       // 0..31
  const int b = blk >> 4;
  const int i0 = (blk & 15) << 4;
  const int tid = threadIdx.x;
  const int lane = tid & 31;
  const int wave = tid >> 5;
  const int col = lane & 15;
  const int half = lane >> 4;

  const int node_i = b * Nn + i0 + col;
  const float* yi = Y + node_i * 16;

  // B operand (U tile, fixed for the block): lane = output column N=col.
  v2f ub[4];
  float uc[8];
  #pragma unroll
  for (int c = 0; c < 4; ++c) {
    ub[c].x = yi[4 * c + 2 * half + 0];
    ub[c].y = yi[4 * c + 2 * half + 1];
  }
  #pragma unroll
  for (int r = 0; r < 8; ++r) uc[r] = yi[half * 8 + r];  // C-layout copy of U

  float acc1[8], acc2[8];
  #pragma unroll
  for (int r = 0; r < 8; ++r) { acc1[r] = 0.0f; acc2[r] = 0.0f; }
  float s1 = 0.0f;

  // 64 iterations per wave; unroll x2 -> two independent load/WMMA streams.
  #pragma unroll 2
  for (int j = wave; j < Nn; j += 4) {
    const float* Mj = Minv + (b * Nn + j) * 256;
    const float* wj = W + (b * Nn + j) * 16 + half * 8;
    const int base = col * 16 + 2 * half;  // A operand: lane = row M=col
    v2f a0, a1, a2, a3;
    a0.x = Mj[base + 0];  a0.y = Mj[base + 1];
    a1.x = Mj[base + 4];  a1.y = Mj[base + 5];
    a2.x = Mj[base + 8];  a2.y = Mj[base + 9];
    a3.x = Mj[base + 12]; a3.y = Mj[base + 13];
    float wr[8];
    #pragma unroll
    for (int r = 0; r < 8; ++r) wr[r] = wj[r];
    const float cj = Cc[b * Nn + j];
    const float bt = beta[(size_t)(b * Nn + i0 + col) * Nn + j];

    v8f p = {0.f, 0.f, 0.f, 0.f, 0.f, 0.f, 0.f, 0.f};
    p = __builtin_amdgcn_wmma_f32_16x16x4_f32(false, a0, false, ub[0], (short)0, p, false, false);
    p = __builtin_amdgcn_wmma_f32_16x16x4_f32(false, a1, false, ub[1], (short)0, p, false, false);
    p = __builtin_amdgcn_wmma_f32_16x16x4_f32(false, a2, false, ub[2], (short)0, p, false, false);
    p = __builtin_amdgcn_wmma_f32_16x16x4_f32(false, a3, false, ub[3], (short)0, p, false, false);

    float qp = 0.0f, tp = 0.0f;
    #pragma unroll
    for (int r = 0; r < 8; ++r) { qp += uc[r] * p[r]; tp += uc[r] * wr[r]; }
    const float q = qp + __shfl_xor(qp, 16, 32);
    const float t = tp + __shfl_xor(tp, 16, 32);
    const float kl = 0.5f * (q - 2.0f * t + cj);
    const float bk = bt * kl;
    #pragma unroll
    for (int r = 0; r < 8; ++r) {
      const float m = p[r] - wr[r];
      acc1[r] += bt * m;
      acc2[r] += bk * m;
    }
    s1 += bk;
  }

  // cross-wave reduction
  __shared__ float red1[32][8];
  __shared__ float red2[32][8];
  __shared__ float redS[32];
  if (tid < 32) {
    redS[tid] = 0.0f;
    #pragma unroll
    for (int r = 0; r < 8; ++r) { red1[tid][r] = 0.0f; red2[tid][r] = 0.0f; }
  }
  __syncthreads();
  #pragma unroll
  for (int r = 0; r < 8; ++r) {
    atomicAdd(&red1[lane][r], acc1[r]);
    atomicAdd(&red2[lane][r], acc2[r]);
  }
  atomicAdd(&redS[lane], s1);
  __syncthreads();

  if (tid < 32) {
    const float s = redS[tid];
    const int c2 = tid & 15, h2 = tid >> 4;
    #pragma unroll
    for (int r = 0; r < 8; ++r) {
      const float A1 = red1[tid][r];
      const float A2 = red2[tid][r];
      const float hv = LAM * A1 + (LAM / KAPPA) * (A2 - s * A1);
      H[(b * Nn + i0 + c2) * 16 + h2 * 8 + r] = hv;
    }
  }
}

// ---------------------------------------------------------------------------
// Kernel 4: finish. grad_mu = ALPHA*(mu_q-mu_p)/sp + E_n @ h_n ; natural grad,
// clip, update; sigma update from self term only. Output stacked [mu; sigma].
// ---------------------------------------------------------------------------
__global__ __launch_bounds__(256) void vfe_finish_kernel(
    const float* __restrict__ mu_q, const float* __restrict__ sigma_q,
    const float* __restrict__ mu_p, const float* __restrict__ sigma_p,
    const float* __restrict__ E, const float* __restrict__ H,
    const float* __restrict__ lr, float* __restrict__ out, int total) {
  const int idx = blockIdx.x * blockDim.x + threadIdx.x;
  if (idx >= total) return;
  const int n = idx >> 4;
  const int k = idx & 15;

  float gm = 0.0f;
  #pragma unroll
  for (int l = 0; l < 16; ++l) gm += E[n * 256 + k * 16 + l] * H[n * 16 + l];

  const float sqs = fmaxf(sigma_q[idx], EPSF);
  const float sps = fmaxf(sigma_p[idx], EPSF);
  const float grad_mu = ALPHA * (mu_q[idx] - mu_p[idx]) / sps + gm;
  float nat_mu = sqs * grad_mu;
  nat_mu = fminf(fmaxf(nat_mu, -GCLIP), GCLIP);
  const float step = lr[0];
  out[idx] = mu_q[idx] - step * nat_mu;

  const float grad_sigma = ALPHA * 0.5f * (1.0f / sps - 1.0f / sqs);
  float nat_sigma = 0.5f * sqs * sqs * grad_sigma;
  nat_sigma = fminf(fmaxf(nat_sigma, -GCLIP), GCLIP);
  out[total + idx] = fmaxf(sigma_q[idx] - step * nat_sigma, EPSF);
}

// ---------------------------------------------------------------------------
// Launch. Workspace layout (floats):
//   E    : [0,        131072)   BN*256
//   Minv : [131072,   262144)   BN*256
//   Y    : [262144,   270336)   BN*16
//   W    : [270336,   278528)   BN*16
//   Cc   : [278528,   279040)   BN
//   H    : [279040,   287232)   BN*16      (~1.15 MB total)
// ---------------------------------------------------------------------------
extern "C" void kernel_launch(void* const* d_in, const int* in_sizes, int n_in,
                              void* d_out, int out_size, void* d_ws, size_t ws_size,
                              hipStream_t stream) {
  const float* mu_q    = (const float*)d_in[0];
  const float* sigma_q = (const float*)d_in[1];
  const float* mu_p    = (const float*)d_in[2];
  const float* sigma_p = (const float*)d_in[3];
  const float* beta    = (const float*)d_in[4];
  const float* phi     = (const float*)d_in[5];
  const float* gen     = (const float*)d_in[6];
  const float* lr      = (const float*)d_in[7];
  float* out = (float*)d_out;

  float* ws = (float*)d_ws;
  float* E    = ws;
  float* Minv = ws + (size_t)BN * 256;
  float* Y    = ws + (size_t)2 * BN * 256;
  float* W    = Y + (size_t)BN * 16;
  float* Cc   = W + (size_t)BN * 16;
  float* H    = Cc + BN;

  vfe_expm_kernel<<<BN, 256, 0, stream>>>(phi, gen, E);
  vfe_prep_kernel<<<BN, 256, 0, stream>>>(sigma_q, mu_q, E, Minv, Y, W, Cc);
  vfe_pairs_kernel<<<Bb * (Nn / 16), 128, 0, stream>>>(Y, W, Cc, Minv, beta, H);
  const int total = Bb * Nn * Kk;
  vfe_finish_kernel<<<(total + 255) / 256, 256, 0, stream>>>(
      mu_q, sigma_q, mu_p, sigma_p, E, H, lr, out, total);
}